// GraphAttention_73632919323214
// MI455X (gfx1250) — compile-verified
//
#include <hip/hip_runtime.h>
#include <hip/hip_bf16.h>

#define BB 8
#define NN 2048
#define FF 512
#define BN (BB * NN)            // 16384 rows total
#define NEGC (-10000.0f)
#define SLOPE 0.2f

typedef float v2f __attribute__((ext_vector_type(2)));
typedef float v8f __attribute__((ext_vector_type(8)));

static __device__ __forceinline__ v8f wmma_f32(v2f a, v2f b, v8f c) {
    // V_WMMA_F32_16X16X4_F32 : D = A(16x4) x B(4x16) + C(16x16), all fp32
    return __builtin_amdgcn_wmma_f32_16x16x4_f32(
        /*neg_a=*/false, a, /*neg_b=*/false, b,
        /*c_mod=*/(short)0, c, /*reuse_a=*/false, /*reuse_b=*/false);
}

// hidden workspace uses a row-pair interleaved layout so that WMMA B-fragments
// (two consecutive K rows at one column) are 8-byte-contiguous in LDS:
//   flat(row, col) = (row & ~1)*FF + col*2 + (row & 1)
static __device__ __forceinline__ size_t hidx(int row, int col) {
    return (size_t)((row & ~1) << 9) + ((unsigned)col << 1) + (row & 1);
}

// ---------------------------------------------------------------------------
// Kernel 1: hidden[BN,FF] = text[BN,FF] @ W[FF,FF] + bias   (interleaved out)
// Block: 256 threads = 8 waves, tile 64 rows x 64 cols; each wave owns two
// independent 16x16 fp32 WMMA accumulator chains.
// ---------------------------------------------------------------------------
__global__ __launch_bounds__(256) void k_hidden_gemm(
    const float* __restrict__ X, const float* __restrict__ W,
    const float* __restrict__ bias, float* __restrict__ H)
{
    __shared__ float ldsA[64 * 33];   // padded stride, A pairs are adjacent
    __shared__ float ldsB[32 * 64];   // k-pair interleaved: (r>>1)*128 + c*2 + (r&1)

    const int tid = threadIdx.x;
    const int w = tid >> 5, l = tid & 31;
    const int wr = w >> 1, wc = w & 1;
    const int m16 = l & 15, half = l >> 4;
    const int rowBase = blockIdx.x * 64;
    const int colBase = blockIdx.y * 64;

    v8f acc0 = {};
    v8f acc1 = {};

    for (int k0 = 0; k0 < FF; k0 += 32) {
        __syncthreads();
        {   // stage A: 64 rows x 32 k
            int r  = tid >> 2;
            int c0 = (tid & 3) * 8;
            const float* src = X + (size_t)(rowBase + r) * FF + k0 + c0;
            float4 v0 = ((const float4*)src)[0];
            float4 v1 = ((const float4*)src)[1];
            float* dst = &ldsA[r * 33 + c0];
            dst[0] = v0.x; dst[1] = v0.y; dst[2] = v0.z; dst[3] = v0.w;
            dst[4] = v1.x; dst[5] = v1.y; dst[6] = v1.z; dst[7] = v1.w;
        }
        {   // stage B: 32 k x 64 cols, k-pair interleaved
            int r  = tid >> 3;
            int c0 = (tid & 7) * 8;
            const float* src = W + (size_t)(k0 + r) * FF + colBase + c0;
            float4 v0 = ((const float4*)src)[0];
            float4 v1 = ((const float4*)src)[1];
            float* dst = &ldsB[(r >> 1) * 128 + (r & 1)];
            dst[(c0 + 0) * 2] = v0.x; dst[(c0 + 1) * 2] = v0.y;
            dst[(c0 + 2) * 2] = v0.z; dst[(c0 + 3) * 2] = v0.w;
            dst[(c0 + 4) * 2] = v1.x; dst[(c0 + 5) * 2] = v1.y;
            dst[(c0 + 6) * 2] = v1.z; dst[(c0 + 7) * 2] = v1.w;
        }
        __syncthreads();
#pragma unroll
        for (int kk = 0; kk < 8; ++kk) {
            const int kb = kk * 4 + 2 * half;     // K pair base (even)
            v2f a;
            a.x = ldsA[(wr * 16 + m16) * 33 + kb];
            a.y = ldsA[(wr * 16 + m16) * 33 + kb + 1];
            const int kp = (kb >> 1) * 128;
            v2f b0 = *(const v2f*)&ldsB[kp + (wc * 32 + m16) * 2];
            v2f b1 = *(const v2f*)&ldsB[kp + (wc * 32 + 16 + m16) * 2];
            acc0 = wmma_f32(a, b0, acc0);
            acc1 = wmma_f32(a, b1, acc1);
        }
    }

    const int col0 = colBase + wc * 32 + m16;
    const float bv0 = bias[col0];
    const float bv1 = bias[col0 + 16];
#pragma unroll
    for (int r = 0; r < 8; ++r) {
        const int row = rowBase + wr * 16 + r + 8 * half;  // C/D: vgpr r -> M=r / M=r+8
        H[hidx(row, col0)]      = acc0[r] + bv0;
        H[hidx(row, col0 + 16)] = acc1[r] + bv1;
    }
}

// ---------------------------------------------------------------------------
// Kernel 2: s[row] = hidden[row,:]·a_src ; d[row] = hidden[row,:]·a_dst
// One wave32 per row, shuffle reduction. (reads the interleaved layout)
// ---------------------------------------------------------------------------
__global__ __launch_bounds__(256) void k_scores(
    const float* __restrict__ H, const float* __restrict__ a_src,
    const float* __restrict__ a_dst, float* __restrict__ sOut,
    float* __restrict__ dOut)
{
    const int w = threadIdx.x >> 5, l = threadIdx.x & 31;
    const int row = blockIdx.x * 8 + w;
    const float* h = H + hidx(row, 0);
    float ss = 0.f, dd = 0.f;
#pragma unroll
    for (int t = 0; t < FF / 32; ++t) {
        const int idx = l + 32 * t;
        const float hv = h[idx << 1];
        ss = fmaf(hv, a_src[idx], ss);
        dd = fmaf(hv, a_dst[idx], dd);
    }
#pragma unroll
    for (int off = 16; off > 0; off >>= 1) {
        ss += __shfl_xor(ss, off, 32);
        dd += __shfl_xor(dd, off, 32);
    }
    if (l == 0) { sOut[row] = ss; dOut[row] = dd; }
}

// ---------------------------------------------------------------------------
// Kernel 3: per-row softmax statistics (online max/sum), one wave32 per row.
// score(i,j) = adj ? NEG : leakyrelu(s_i + d_j). All-masked rows naturally
// produce max=NEG, sum=N -> uniform weights, matching the reference softmax.
// ---------------------------------------------------------------------------
__global__ __launch_bounds__(256) void k_rowstats(
    const int* __restrict__ adj, const float* __restrict__ sArr,
    const float* __restrict__ dArr, float* __restrict__ rmax,
    float* __restrict__ rsum)
{
    const int w = threadIdx.x >> 5, l = threadIdx.x & 31;
    const int row = blockIdx.x * 8 + w;          // 0..BN-1
    const int bb = row >> 11;                    // row / NN
    const int4*   adjRow = (const int4*)(adj + (size_t)row * NN);
    const float4* dRow   = (const float4*)(dArr + bb * NN);
    const float si = sArr[row];

    float m = -3.0e38f, sum = 0.f;
#define LRELU(v) ((v) >= 0.f ? (v) : SLOPE * (v))
#define ONLINE(v)                                              \
    do {                                                       \
        float _v = (v);                                        \
        if (_v > m) { sum = sum * __expf(m - _v) + 1.f; m = _v; } \
        else        { sum += __expf(_v - m); }                 \
    } while (0)

#pragma unroll 4
    for (int t = 0; t < NN / 128; ++t) {
        const int idx = l + 32 * t;
        __builtin_prefetch(&adjRow[idx + 64], 0, 1);   // stream adj ahead (HBM)
        const int4   a4 = adjRow[idx];
        const float4 d4 = dRow[idx];
        float v;
        v = LRELU(si + d4.x); if (a4.x) v = NEGC; ONLINE(v);
        v = LRELU(si + d4.y); if (a4.y) v = NEGC; ONLINE(v);
        v = LRELU(si + d4.z); if (a4.z) v = NEGC; ONLINE(v);
        v = LRELU(si + d4.w); if (a4.w) v = NEGC; ONLINE(v);
    }
#pragma unroll
    for (int off = 16; off > 0; off >>= 1) {
        const float m2 = __shfl_xor(m, off, 32);
        const float s2 = __shfl_xor(sum, off, 32);
        const float nm = fmaxf(m, m2);
        sum = sum * __expf(m - nm) + s2 * __expf(m2 - nm);
        m = nm;
    }
    if (l == 0) { rmax[row] = m; rsum[row] = sum; }
#undef ONLINE
#undef LRELU
}

// ---------------------------------------------------------------------------
// Kernel 4: out[b,i,:] = (1/rsum_i) * sum_j exp(score(i,j)-rmax_i) * hidden[b,j,:]
// Block: 1024 threads = 32 waves; tile = 64 rows x full 512 cols (halves the
// L2 re-read of hidden vs 32-row tiles). Wave (rg,fg): rg in 0..3 -> 16-row
// strip, fg in 0..7 -> 64-col strip (4 WMMA accumulators). Per 16-wide j
// tile: stage hidden[16x512] (contiguous 32KB copy; global layout already
// pair-interleaved so B-fragments are contiguous) + P[64x16] in LDS, then
// each wave issues 16 fp32 WMMAs with single-ds_load_b64 B fragments.
// ---------------------------------------------------------------------------
__global__ __launch_bounds__(1024) void k_attn_out(
    const int* __restrict__ adj, const float* __restrict__ H,
    const float* __restrict__ sArr, const float* __restrict__ dArr,
    const float* __restrict__ rmax, const float* __restrict__ rsum,
    float* __restrict__ out)
{
    __shared__ float ldsH[16 * FF];    // 32 KB, pair-interleaved j-tile
    __shared__ float ldsP[64 * 17];    // exp-weights, padded stride
    __shared__ float ldsInv[64];

    const int tid = threadIdx.x;
    const int w = tid >> 5, l = tid & 31;
    const int rg = w >> 3, fg = w & 7;
    const int m16 = l & 15, half = l >> 4;
    const int bb = blockIdx.y;
    const int i0 = blockIdx.x * 64;
    const int bn = bb * NN;

    // fixed per-thread coords for the P (weights) tile: one entry per thread
    const int pii = tid >> 4;          // 0..63 (row within tile)
    const int pjj = tid & 15;          // 0..15 (col within j tile)
    const float psi = sArr[bn + i0 + pii];
    const float prm = rmax[bn + i0 + pii];
    const int*   adjP = adj + (size_t)(bn + i0 + pii) * NN + pjj;
    const float* dP   = dArr + bn + pjj;

    if (tid < 64) ldsInv[tid] = 1.0f / rsum[bn + i0 + tid];

    v8f acc[4] = {};

    for (int j0 = 0; j0 < NN; j0 += 16) {
        __syncthreads();
        {   // stage hidden tile: 16 rows x 512 cols, contiguous 32 KB copy
            const float4* hsrc = (const float4*)(H + (size_t)(bn + j0) * FF);
            float4* hdst = (float4*)ldsH;
#pragma unroll
            for (int kq = 0; kq < 2; ++kq)
                hdst[tid + 1024 * kq] = hsrc[tid + 1024 * kq];
        }
        {   // stage P tile: exp(score - rowmax); prefetch next adj tile (HBM)
            __builtin_prefetch(adjP + j0 + 16, 0, 3);
            const int av = adjP[j0];
            float v = psi + dP[j0];
            v = v >= 0.f ? v : SLOPE * v;
            if (av) v = NEGC;
            ldsP[pii * 17 + pjj] = __expf(v - prm);
        }
        __syncthreads();

        v2f a[4];
#pragma unroll
        for (int kk = 0; kk < 4; ++kk) {
            const int kb = kk * 4 + 2 * half;
            a[kk].x = ldsP[(rg * 16 + m16) * 17 + kb];
            a[kk].y = ldsP[(rg * 16 + m16) * 17 + kb + 1];
        }
#pragma unroll
        for (int nt = 0; nt < 4; ++nt) {
            const int col = fg * 64 + nt * 16 + m16;
#pragma unroll
            for (int kk = 0; kk < 4; ++kk) {
                const int kp = (kk * 2 + half) * 1024;   // (kb>>1)*2*FF
                v2f b = *(const v2f*)&ldsH[kp + col * 2];
                acc[nt] = wmma_f32(a[kk], b, acc[nt]);
            }
        }
    }

#pragma unroll
    for (int nt = 0; nt < 4; ++nt) {
        const int col = fg * 64 + nt * 16 + m16;
#pragma unroll
        for (int r = 0; r < 8; ++r) {
            const int m = rg * 16 + r + 8 * half;
            out[(size_t)(bn + i0 + m) * FF + col] = acc[nt][r] * ldsInv[m];
        }
    }
}

// ---------------------------------------------------------------------------
extern "C" void kernel_launch(void* const* d_in, const int* in_sizes, int n_in,
                              void* d_out, int out_size, void* d_ws, size_t ws_size,
                              hipStream_t stream) {
    const float* text  = (const float*)d_in[0];   // [8,2048,512] f32
    const int*   adj   = (const int*)  d_in[1];   // [8,2048,2048] i32
    const float* W     = (const float*)d_in[2];   // [512,512]
    const float* bias  = (const float*)d_in[3];   // [512]
    const float* a_src = (const float*)d_in[4];   // [512]
    const float* a_dst = (const float*)d_in[5];   // [512]
    float* out = (float*)d_out;                   // [8,2048,512] f32

    // workspace layout (floats): hidden[BN*FF] | s[BN] | d[BN] | rmax[BN] | rsum[BN]
    float* H     = (float*)d_ws;
    float* sArr  = H + (size_t)BN * FF;
    float* dArr  = sArr + BN;
    float* rmaxA = dArr + BN;
    float* rsumA = rmaxA + BN;

    k_hidden_gemm<<<dim3(BN / 64, FF / 64), 256, 0, stream>>>(text, W, bias, H);
    k_scores    <<<dim3(BN / 8), 256, 0, stream>>>(H, a_src, a_dst, sArr, dArr);
    k_rowstats  <<<dim3(BN / 8), 256, 0, stream>>>(adj, sArr, dArr, rmaxA, rsumA);
    k_attn_out  <<<dim3(NN / 64, BB), 1024, 0, stream>>>(adj, H, sArr, dArr, rmaxA,
                                                         rsumA, out);
}